// Model_30305289241051
// MI455X (gfx1250) — compile-verified
//
#include <hip/hip_runtime.h>
#include <hip/hip_bf16.h>
#include <cstdint>

// ---------------------------------------------------------------------------
// GIN link-predictor for MI455X (gfx1250, wave32).
// GEMMs on the matrix pipe via V_WMMA_F32_16X16X4_F32 (full fp32).
// Each wave owns a 16x64 output strip (4 accumulators) so one A fetch feeds
// four WMMAs. Scatter-add / BN / edge-dot are shaped around the 192MB L2.
// ---------------------------------------------------------------------------

typedef __attribute__((ext_vector_type(2))) float v2f;
typedef __attribute__((ext_vector_type(8))) float v8f;

#define D 128  // hidden dim (fixed by the model)

#define WMMA_F32(a, b, c)                                                  \
  __builtin_amdgcn_wmma_f32_16x16x4_f32(false, (a), false, (b), (short)0,  \
                                        (c), false, false)

// out[M x 128] = act( (A [+ A2]) @ W + bias ), W 128x128 row-major.
// Block = 64 threads = 2 waves; wave w computes rows m0..m0+15, cols
// 64w..64w+63 (4 x 16x16 WMMA tiles). gridDim.x = ceil(M/16).
template <bool HAS_A2, bool RELU>
__global__ __launch_bounds__(64) void gemm128_wmma_kernel(
    const float* __restrict__ A, const float* __restrict__ A2,
    const float* __restrict__ W, const float* __restrict__ bias,
    float* __restrict__ out, int M) {
  const int lane  = threadIdx.x & 31;
  const int wave  = threadIdx.x >> 5;  // 0..1
  const int half  = lane >> 4;         // 0: lanes 0-15, 1: lanes 16-31
  const int l16   = lane & 15;
  const int m0    = blockIdx.x * 16;
  const int nbase = wave * 64;

  // Clamp A row so all 32 lanes stay active (EXEC must be all-ones at WMMA).
  int arow = m0 + l16;
  if (arow >= M) arow = M - 1;
  const float* __restrict__ Arow  = A + (size_t)arow * D;
  const float* __restrict__ A2row = HAS_A2 ? (A2 + (size_t)arow * D) : nullptr;

  // Pre-load bias into accumulators: acc register r of tile t holds
  // (row m0+r+8*half, col nbase+16t+l16) -> bias depends only on the column.
  const float bv0 = bias[nbase + 0 + l16];
  const float bv1 = bias[nbase + 16 + l16];
  const float bv2 = bias[nbase + 32 + l16];
  const float bv3 = bias[nbase + 48 + l16];
  v8f acc0, acc1, acc2, acc3;
#pragma unroll
  for (int r = 0; r < 8; ++r) {
    acc0[r] = bv0; acc1[r] = bv1; acc2[r] = bv2; acc3[r] = bv3;
  }

#pragma unroll 2
  for (int k0 = 0; k0 < D; k0 += 4) {
    const int ka = k0 + half * 2;  // lanes 0-15: K={k0,k0+1}; 16-31: K={k0+2,k0+3}
    v2f a;
    a.x = Arow[ka];
    a.y = Arow[ka + 1];
    if (HAS_A2) {
      a.x += A2row[ka];
      a.y += A2row[ka + 1];
    }
    const float* __restrict__ Wk0 = W + (size_t)ka * D + l16;
    const float* __restrict__ Wk1 = Wk0 + D;
    v2f b0 = {Wk0[nbase +  0], Wk1[nbase +  0]};
    v2f b1 = {Wk0[nbase + 16], Wk1[nbase + 16]};
    v2f b2 = {Wk0[nbase + 32], Wk1[nbase + 32]};
    v2f b3 = {Wk0[nbase + 48], Wk1[nbase + 48]};
    acc0 = WMMA_F32(a, b0, acc0);
    acc1 = WMMA_F32(a, b1, acc1);
    acc2 = WMMA_F32(a, b2, acc2);
    acc3 = WMMA_F32(a, b3, acc3);
  }

  if (m0 + 16 <= M) {  // full tile: unpredicated stores
#pragma unroll
    for (int r = 0; r < 8; ++r) {
      float* __restrict__ o = out + (size_t)(m0 + r + 8 * half) * D + l16;
      float v0 = acc0[r], v1 = acc1[r], v2 = acc2[r], v3 = acc3[r];
      if (RELU) {
        v0 = fmaxf(v0, 0.0f); v1 = fmaxf(v1, 0.0f);
        v2 = fmaxf(v2, 0.0f); v3 = fmaxf(v3, 0.0f);
      }
      o[nbase + 0] = v0; o[nbase + 16] = v1;
      o[nbase + 32] = v2; o[nbase + 48] = v3;
    }
  } else {  // generic tail
#pragma unroll
    for (int r = 0; r < 8; ++r) {
      const int row = m0 + r + 8 * half;
      if (row < M) {
        float* __restrict__ o = out + (size_t)row * D + l16;
        float v0 = acc0[r], v1 = acc1[r], v2 = acc2[r], v3 = acc3[r];
        if (RELU) {
          v0 = fmaxf(v0, 0.0f); v1 = fmaxf(v1, 0.0f);
          v2 = fmaxf(v2, 0.0f); v3 = fmaxf(v3, 0.0f);
        }
        o[nbase + 0] = v0; o[nbase + 16] = v1;
        o[nbase + 32] = v2; o[nbase + 48] = v3;
      }
    }
  }
}

// agg[dst[e]][:] += h[src[e]][:]  -- 32 lanes per edge, float4 per lane.
__global__ __launch_bounds__(256) void scatter_add_kernel(
    const float* __restrict__ h, const int* __restrict__ src,
    const int* __restrict__ dst, float* __restrict__ agg, int E) {
  const long long idx = (long long)blockIdx.x * blockDim.x + threadIdx.x;
  const long long total = (long long)E * 32;
  if (idx >= total) return;
  const int e = (int)(idx >> 5);
  const int l = (int)(idx & 31);
  const int s = src[e];
  const int d = dst[e];
  const float4 v = *(const float4*)(h + (size_t)s * D + l * 4);
  float* ad = agg + (size_t)d * D + l * 4;
  atomicAdd(ad + 0, v.x);
  atomicAdd(ad + 1, v.y);
  atomicAdd(ad + 2, v.z);
  atomicAdd(ad + 3, v.w);
}

// Per-feature sum / sum-of-squares partials; one atomic per feature per block.
__global__ __launch_bounds__(128) void bn_stats_kernel(
    const float* __restrict__ z, float* __restrict__ stats, int M,
    int rowsPerBlock) {
  const int f  = threadIdx.x;  // feature 0..127; coalesced across a row
  const int r0 = blockIdx.x * rowsPerBlock;
  int r1 = r0 + rowsPerBlock;
  if (r1 > M) r1 = M;
  float s = 0.0f, ss = 0.0f;
  for (int r = r0; r < r1; ++r) {
    const float v = z[(size_t)r * D + f];
    s += v;
    ss += v * v;
  }
  atomicAdd(&stats[f], s);
  atomicAdd(&stats[D + f], ss);
}

// scale = gamma * rsqrt(var + eps); shift = beta - mu * scale
__global__ void bn_finalize_kernel(const float* __restrict__ stats,
                                   const float* __restrict__ gamma,
                                   const float* __restrict__ beta,
                                   float* __restrict__ ssh, int M) {
  const int f = threadIdx.x;
  const float inv_n = 1.0f / (float)M;
  const float mu  = stats[f] * inv_n;
  const float var = stats[D + f] * inv_n - mu * mu;
  const float sc  = gamma[f] * rsqrtf(var + 1e-5f);
  ssh[f]     = sc;
  ssh[D + f] = beta[f] - mu * sc;
}

// z = relu(z * scale[f] + shift[f]), in place.
__global__ __launch_bounds__(256) void bn_apply_relu_kernel(
    float* __restrict__ z, const float* __restrict__ ssh, long long total) {
  const long long i = (long long)blockIdx.x * blockDim.x + threadIdx.x;
  if (i >= total) return;
  const int f = (int)(i & (D - 1));
  z[i] = fmaxf(fmaf(z[i], ssh[f], ssh[D + f]), 0.0f);
}

// pred[e] = dot(h[ea[e]], h[eb[e]]) over 128 dims; one wave32 per edge,
// one b128 load per endpoint per lane, then a 5-step wave32 xor-reduction.
__global__ __launch_bounds__(256) void edge_dot_kernel(
    const float* __restrict__ h, const int* __restrict__ ea,
    const int* __restrict__ eb, float* __restrict__ pred, int EL) {
  const int widx = (int)(((long long)blockIdx.x * blockDim.x + threadIdx.x) >> 5);
  const int lane = threadIdx.x & 31;
  if (widx >= EL) return;  // wave-uniform exit
  const float4 va = *(const float4*)(h + (size_t)ea[widx] * D + lane * 4);
  const float4 vb = *(const float4*)(h + (size_t)eb[widx] * D + lane * 4);
  float acc = va.x * vb.x;
  acc = fmaf(va.y, vb.y, acc);
  acc = fmaf(va.z, vb.z, acc);
  acc = fmaf(va.w, vb.w, acc);
#pragma unroll
  for (int off = 16; off; off >>= 1) acc += __shfl_xor(acc, off, 32);
  if (lane == 0) pred[widx] = acc;
}

extern "C" void kernel_launch(void* const* d_in, const int* in_sizes, int n_in,
                              void* d_out, int out_size, void* d_ws,
                              size_t ws_size, hipStream_t stream) {
  const float* x     = (const float*)d_in[0];
  const float* lin_w = (const float*)d_in[1];
  const float* lin_b = (const float*)d_in[2];
  const float* W1    = (const float*)d_in[3];
  const float* b1    = (const float*)d_in[4];
  const float* gamma = (const float*)d_in[5];
  const float* beta  = (const float*)d_in[6];
  const float* W2    = (const float*)d_in[7];
  const float* b2    = (const float*)d_in[8];
  const int* eidx    = (const int*)d_in[9];
  const int* elidx   = (const int*)d_in[10];
  float* pred = (float*)d_out;

  const int N  = in_sizes[0] / D;
  const int E  = in_sizes[9] / 2;
  const int EL = in_sizes[10] / 2;
  const size_t ND = (size_t)N * D;

  float* h     = (float*)d_ws;
  float* agg   = h + ND;
  float* z     = agg + ND;
  float* stats = z + ND;      // 2*D floats
  float* ssh   = stats + 2 * D;

  const int mTiles = (N + 15) / 16;
  const dim3 gGrid(mTiles), gBlk(64);

  // h = x @ lin_w + lin_b
  gemm128_wmma_kernel<false, false><<<gGrid, gBlk, 0, stream>>>(
      x, nullptr, lin_w, lin_b, h, N);

  const long long sTotal = (long long)E * 32;
  const int sBlocks = (int)((sTotal + 255) / 256);
  const int rowsPerBlock = 256;
  const int statBlocks = (N + rowsPerBlock - 1) / rowsPerBlock;
  const long long elemTotal = (long long)N * D;
  const int applyBlocks = (int)((elemTotal + 255) / 256);

  for (int l = 0; l < 3; ++l) {
    const float* W1l = W1 + (size_t)l * D * D;
    const float* W2l = W2 + (size_t)l * D * D;
    // agg = segment_sum(h[src], dst)
    hipMemsetAsync(agg, 0, ND * sizeof(float), stream);
    scatter_add_kernel<<<sBlocks, 256, 0, stream>>>(h, eidx, eidx + E, agg, E);
    // z = (h + agg) @ W1[l] + b1[l]
    gemm128_wmma_kernel<true, false><<<gGrid, gBlk, 0, stream>>>(
        h, agg, W1l, b1 + l * D, z, N);
    // batch-norm (training-mode batch stats) + relu
    hipMemsetAsync(stats, 0, 2 * D * sizeof(float), stream);
    bn_stats_kernel<<<statBlocks, 128, 0, stream>>>(z, stats, N, rowsPerBlock);
    bn_finalize_kernel<<<1, D, 0, stream>>>(stats, gamma + l * D, beta + l * D,
                                            ssh, N);
    bn_apply_relu_kernel<<<applyBlocks, 256, 0, stream>>>(z, ssh, elemTotal);
    // h = relu(z @ W2[l] + b2[l])
    gemm128_wmma_kernel<false, true><<<gGrid, gBlk, 0, stream>>>(
        z, nullptr, W2l, b2 + l * D, h, N);
  }

  // pred[e] = <h[a], h[b]>
  const long long dTotal = (long long)EL * 32;
  edge_dot_kernel<<<(int)((dTotal + 255) / 256), 256, 0, stream>>>(
      h, elidx, elidx + EL, pred, EL);
}